// Cat_Linear_Encoder_51376398794911
// MI455X (gfx1250) — compile-verified
//
#include <hip/hip_runtime.h>
#include <hip/hip_bf16.h>
#include <math.h>

// N=2048, D=64, H=64 fixed by the reference.
#define NROWS 2048
#define DDIM  64
#define HDIM  64

typedef __attribute__((ext_vector_type(2))) float v2f;
typedef __attribute__((ext_vector_type(8))) float v8f;

// ---------------------------------------------------------------------------
// Kernel A: G[2048][128] = z[2048][64] @ M[64][128]
//   columns 0..63   : hi[i][h] = sum_d z[i][d] * W1[h][d]
//   columns 64..127 : hj[i][h] = sum_d z[i][d] * W1[h][64+d]
// One wave32 per 16x16 tile of G; K=64 via 16 chained V_WMMA_F32_16X16X4_F32.
// ---------------------------------------------------------------------------
__global__ __launch_bounds__(32) void gemm_hij_wmma(const float* __restrict__ z,
                                                    const float* __restrict__ W1,
                                                    float* __restrict__ G) {
  const int lane   = threadIdx.x;          // 0..31, full wave (EXEC all 1s)
  const int tile_m = blockIdx.x;           // 0..127 : 16 rows of z each
  const int tile_n = blockIdx.y;           // 0..7   : 16 cols of G each
  const int l15    = lane & 15;
  const int khi    = (lane >> 4) * 2;      // lanes 0-15 -> K {0,1}; 16-31 -> K {2,3}

  const int m = tile_m * 16 + l15;         // A-matrix row this lane holds
  const int n = tile_n * 16 + l15;         // B-matrix column this lane holds

  v8f c = {};                              // 16x16 f32 accumulator (8 VGPRs)

  #pragma unroll
  for (int kk = 0; kk < 16; ++kk) {        // K blocks of 4
    const int k0 = kk * 4 + khi;
    const int k1 = k0 + 1;

    // A fragment: 16x4 f32, ISA layout (VGPR0 = K even, VGPR1 = K odd per half-wave)
    v2f a;
    a.x = z[m * DDIM + k0];
    a.y = z[m * DDIM + k1];

    // B fragment: 4x16 f32, row K striped across lanes.
    // B[k][n] = (n < 64) ? W1[n][k] : W1[n-64][64+k]   (the concatenated Wa^T|Wb^T)
    v2f b;
    b.x = (n < HDIM) ? W1[n * (2 * DDIM) + k0]
                     : W1[(n - HDIM) * (2 * DDIM) + DDIM + k0];
    b.y = (n < HDIM) ? W1[n * (2 * DDIM) + k1]
                     : W1[(n - HDIM) * (2 * DDIM) + DDIM + k1];

    // D = A*B + C  (8 args: neg_a, A, neg_b, B, c_mod, C, reuse_a, reuse_b)
    c = __builtin_amdgcn_wmma_f32_16x16x4_f32(false, a, false, b,
                                              (short)0, c, false, false);
  }

  // C/D layout: VGPR r -> (M=r, N=lane) for lanes 0-15, (M=r+8, N=lane-16) for 16-31
  const int rbase = tile_m * 16 + ((lane >> 4) ? 8 : 0);
  const int col   = tile_n * 16 + l15;
  #pragma unroll
  for (int r = 0; r < 8; ++r) {
    G[(rbase + r) * (2 * HDIM) + col] = c[r];
  }
}

// ---------------------------------------------------------------------------
// Kernel B: out[i][j] = sigmoid( sum_h relu(hi[i][h]+hj[j][h]+b1[h]) * W2[h] + b2 )
//           with zeroed diagonal.  16x16 output tile per 256-thread block.
// Store-bandwidth bound (16.8 MB out); hi/hj tiles staged in LDS (8 KB reuse).
// ---------------------------------------------------------------------------
__global__ __launch_bounds__(256) void pairwise_score(const float* __restrict__ G,
                                                      const float* __restrict__ b1,
                                                      const float* __restrict__ W2,
                                                      const float* __restrict__ b2,
                                                      float* __restrict__ out) {
  __shared__ float sHi[16][HDIM + 1];   // +1 pad: avoid 16-way LDS bank conflicts
  __shared__ float sHj[16][HDIM + 1];
  __shared__ float sW2[HDIM];

  const int ti = blockIdx.y;            // i tile (rows)
  const int tj = blockIdx.x;            // j tile (cols)
  const int t  = threadIdx.x;           // 0..255

  for (int v = t; v < 16 * HDIM; v += 256) {
    const int r = v >> 6;               // 0..15
    const int h = v & (HDIM - 1);       // 0..63
    sHi[r][h] = G[(ti * 16 + r) * (2 * HDIM) + h];
    sHj[r][h] = G[(tj * 16 + r) * (2 * HDIM) + HDIM + h] + b1[h];
  }
  if (t < HDIM) sW2[t] = W2[t];
  __syncthreads();

  const int li = t >> 4;                // 0..15 local i
  const int lj = t & 15;                // 0..15 local j
  const float* hi = sHi[li];
  const float* hj = sHj[lj];

  float acc = 0.0f;
  #pragma unroll
  for (int h = 0; h < HDIM; ++h) {
    const float x = hi[h] + hj[h];
    acc = fmaf(fmaxf(x, 0.0f), sW2[h], acc);
  }
  acc += b2[0];

  const float p = 1.0f / (1.0f + __expf(-acc));
  const int gi = ti * 16 + li;
  const int gj = tj * 16 + lj;
  out[gi * NROWS + gj] = (gi == gj) ? 0.0f : p;
}

// ---------------------------------------------------------------------------
extern "C" void kernel_launch(void* const* d_in, const int* in_sizes, int n_in,
                              void* d_out, int out_size, void* d_ws, size_t ws_size,
                              hipStream_t stream) {
  const float* z  = (const float*)d_in[0];   // (2048, 64)
  const float* W1 = (const float*)d_in[1];   // (64, 128)
  const float* b1 = (const float*)d_in[2];   // (64,)
  const float* W2 = (const float*)d_in[3];   // (1, 64)
  const float* b2 = (const float*)d_in[4];   // (1,)
  float* out = (float*)d_out;                // (2048, 2048)
  float* G   = (float*)d_ws;                 // 2048 x 128 f32 = 1 MB scratch

  dim3 gridA(NROWS / 16, (2 * HDIM) / 16);   // 128 x 8 tiles, one wave each
  gemm_hij_wmma<<<gridA, 32, 0, stream>>>(z, W1, G);

  dim3 gridB(NROWS / 16, NROWS / 16);        // 128 x 128 output tiles
  pairwise_score<<<gridB, 256, 0, stream>>>(G, b1, W2, b2, out);
}